// GraphSAGE_53257594470606
// MI455X (gfx1250) — compile-verified
//
#include <hip/hip_runtime.h>

// GraphSAGE (3 layers, d=64) for gfx1250.
// Edge scatter via native f32 global atomics (L2-resident), node GEMMs via
// V_WMMA_F32_16X16X4_F32 (fp32-exact, wave32, 16x16 C tiles).

typedef float v2f __attribute__((ext_vector_type(2)));
typedef float v8f __attribute__((ext_vector_type(8)));

#define DFEAT 64

__device__ __forceinline__ v8f wmma4(v2f a, v2f b, v8f c) {
  // 8 args: (neg_a, A, neg_b, B, c_mod, C, reuse_a, reuse_b)
  return __builtin_amdgcn_wmma_f32_16x16x4_f32(false, a, false, b, (short)0, c,
                                               false, false);
}

__global__ void fill_zero_kernel(float4* __restrict__ p, int n4) {
  int i = blockIdx.x * blockDim.x + threadIdx.x;
  if (i < n4) p[i] = make_float4(0.f, 0.f, 0.f, 0.f);
}

__global__ void count_kernel(const long long* __restrict__ dst,
                             float* __restrict__ cnt, int E) {
  int e = blockIdx.x * blockDim.x + threadIdx.x;
  if (e < E) unsafeAtomicAdd(cnt + (int)dst[e], 1.0f);
}

__global__ void inv_kernel(float* __restrict__ cnt, int n) {
  int i = blockIdx.x * blockDim.x + threadIdx.x;
  if (i < n) cnt[i] = 1.0f / fmaxf(cnt[i], 1.0f);
}

// 16 lanes per edge; each lane moves 4 consecutive floats of the 64-f row.
__global__ void scatter_kernel(const float* __restrict__ x,
                               const long long* __restrict__ src,
                               const long long* __restrict__ dst,
                               float* __restrict__ agg, int E) {
  int t = blockIdx.x * blockDim.x + threadIdx.x;
  int e = t >> 4;
  if (e >= E) return;
  int q = t & 15;
  int s = (int)src[e];
  int d = (int)dst[e];
  const float4 v = *(const float4*)(x + s * DFEAT + q * 4);
  float* p = agg + d * DFEAT + q * 4;
  unsafeAtomicAdd(p + 0, v.x);
  unsafeAtomicAdd(p + 1, v.y);
  unsafeAtomicAdd(p + 2, v.z);
  unsafeAtomicAdd(p + 3, v.w);
}

// out[tile rows, 0:64] = relu_opt( (agg*inv) @ Wl + b + xin @ Wr )
// One wave = one 16-node tile; 4 C tiles of 16x16 f32 (8 VGPRs each).
__global__ __launch_bounds__(256) void sage_gemm_kernel(
    const float* __restrict__ xin, const float* __restrict__ agg,
    const float* __restrict__ inv, const float* __restrict__ Wl,
    const float* __restrict__ Wr, const float* __restrict__ bias,
    float* __restrict__ out, int ntiles, int relu) {
  int tile = blockIdx.x * (blockDim.x >> 5) + (threadIdx.x >> 5);
  if (tile >= ntiles) return;  // wave-uniform: EXEC all-1 for WMMA
  int lane = threadIdx.x & 31;
  int row = lane & 15;          // A: M index / B: N index (per ISA layout)
  int kh = (lane >> 4) << 1;    // lanes 16-31 hold K+2,K+3
  int node = tile * 16 + row;
  float scale = inv[node];

  v8f c0 = {}, c1 = {}, c2 = {}, c3 = {};

  const float* ap = agg + node * DFEAT + kh;
  const float* xp = xin + node * DFEAT + kh;

  // GEMM 1: mean-neighbors x Wl  (scale folded into A fragment)
#pragma unroll
  for (int k0 = 0; k0 < DFEAT; k0 += 4) {
    v2f a;
    a.x = ap[k0] * scale;
    a.y = ap[k0 + 1] * scale;
    const float* w = Wl + (k0 + kh) * DFEAT + row;
    v2f b0, b1, b2, b3;
    b0.x = w[0];  b0.y = w[DFEAT + 0];
    b1.x = w[16]; b1.y = w[DFEAT + 16];
    b2.x = w[32]; b2.y = w[DFEAT + 32];
    b3.x = w[48]; b3.y = w[DFEAT + 48];
    c0 = wmma4(a, b0, c0);
    c1 = wmma4(a, b1, c1);
    c2 = wmma4(a, b2, c2);
    c3 = wmma4(a, b3, c3);
  }
  // GEMM 2: self-features x Wr, accumulate into same C
#pragma unroll
  for (int k0 = 0; k0 < DFEAT; k0 += 4) {
    v2f a;
    a.x = xp[k0];
    a.y = xp[k0 + 1];
    const float* w = Wr + (k0 + kh) * DFEAT + row;
    v2f b0, b1, b2, b3;
    b0.x = w[0];  b0.y = w[DFEAT + 0];
    b1.x = w[16]; b1.y = w[DFEAT + 16];
    b2.x = w[32]; b2.y = w[DFEAT + 32];
    b3.x = w[48]; b3.y = w[DFEAT + 48];
    c0 = wmma4(a, b0, c0);
    c1 = wmma4(a, b1, c1);
    c2 = wmma4(a, b2, c2);
    c3 = wmma4(a, b3, c3);
  }

  // Epilogue: C layout = VGPR v -> M = v + 8*(lane>=16), N = lane&15
  int colg = row;
  int mbase = tile * 16 + ((lane >> 4) << 3);
  float bb0 = bias[colg], bb1 = bias[16 + colg];
  float bb2 = bias[32 + colg], bb3 = bias[48 + colg];
#pragma unroll
  for (int v = 0; v < 8; ++v) {
    float o0 = c0[v] + bb0, o1 = c1[v] + bb1;
    float o2 = c2[v] + bb2, o3 = c3[v] + bb3;
    if (relu) {
      o0 = fmaxf(o0, 0.f); o1 = fmaxf(o1, 0.f);
      o2 = fmaxf(o2, 0.f); o3 = fmaxf(o3, 0.f);
    }
    float* po = out + (size_t)(mbase + v) * DFEAT;
    po[colg] = o0;
    po[16 + colg] = o1;
    po[32 + colg] = o2;
    po[48 + colg] = o3;
  }
}

extern "C" void kernel_launch(void* const* d_in, const int* in_sizes, int n_in,
                              void* d_out, int out_size, void* d_ws,
                              size_t ws_size, hipStream_t stream) {
  const float* x        = (const float*)d_in[0];
  const long long* ei   = (const long long*)d_in[1];  // int64 [2, E]
  const float* Wl1 = (const float*)d_in[2];
  const float* Wr1 = (const float*)d_in[3];
  const float* b1  = (const float*)d_in[4];
  const float* Wl2 = (const float*)d_in[5];
  const float* Wr2 = (const float*)d_in[6];
  const float* b2  = (const float*)d_in[7];
  const float* Wl3 = (const float*)d_in[8];
  const float* Wr3 = (const float*)d_in[9];
  const float* b3  = (const float*)d_in[10];

  const int N = in_sizes[0] / DFEAT;  // 50000
  const int E = in_sizes[1] / 2;      // 800000
  const long long* src  = ei;
  const long long* dstE = ei + E;

  float* ws  = (float*)d_ws;
  float* cnt = ws;                          // N floats (becomes 1/max(cnt,1))
  float* agg = ws + ((N + 3) & ~3);         // N*64
  float* h1  = agg + (size_t)N * DFEAT;     // N*64
  float* h2  = h1 + (size_t)N * DFEAT;      // N*64

  const int B = 256;
  const int nf4 = N * DFEAT / 4;
  const int ntiles = N / 16;
  const int gemmBlocks = (ntiles + 7) / 8;  // 8 waves per 256-thread block
  const int scatBlocks = (E * 16 + B - 1) / B;

  // degree counts (same edge list every layer) -> reciprocal
  fill_zero_kernel<<<(N / 4 + B - 1) / B, B, 0, stream>>>((float4*)cnt, N / 4);
  count_kernel<<<(E + B - 1) / B, B, 0, stream>>>(dstE, cnt, E);
  inv_kernel<<<(N + B - 1) / B, B, 0, stream>>>(cnt, N);

  // layer 1 (+ReLU)
  fill_zero_kernel<<<(nf4 + B - 1) / B, B, 0, stream>>>((float4*)agg, nf4);
  scatter_kernel<<<scatBlocks, B, 0, stream>>>(x, src, dstE, agg, E);
  sage_gemm_kernel<<<gemmBlocks, B, 0, stream>>>(x, agg, cnt, Wl1, Wr1, b1, h1,
                                                 ntiles, 1);
  // layer 2 (+ReLU)
  fill_zero_kernel<<<(nf4 + B - 1) / B, B, 0, stream>>>((float4*)agg, nf4);
  scatter_kernel<<<scatBlocks, B, 0, stream>>>(h1, src, dstE, agg, E);
  sage_gemm_kernel<<<gemmBlocks, B, 0, stream>>>(h1, agg, cnt, Wl2, Wr2, b2, h2,
                                                 ntiles, 1);
  // layer 3 (no activation) -> d_out
  fill_zero_kernel<<<(nf4 + B - 1) / B, B, 0, stream>>>((float4*)agg, nf4);
  scatter_kernel<<<scatBlocks, B, 0, stream>>>(h2, src, dstE, agg, E);
  sage_gemm_kernel<<<gemmBlocks, B, 0, stream>>>(h2, agg, cnt, Wl3, Wr3, b3,
                                                 (float*)d_out, ntiles, 0);
}